// LRU_33011118637310
// MI455X (gfx1250) — compile-verified
//
#include <hip/hip_runtime.h>

// LRU (diagonal complex linear recurrent unit), L=32768, D_MODEL=D_HIDDEN=256.
// MI455X plan:
//  - fp32 WMMA (v_wmma_f32_16x16x4_f32) for the two complex GEMMs.
//  - Weight panels (64 cols x 256 K x 2 planes = 128 KB) DMA'd into LDS once
//    per block via the Tensor Data Mover (tensor_load_to_lds + s_wait_tensorcnt),
//    using TDM padding (260-float row stride) for conflict-free ds_load_b64.
//  - 3-phase blocked scan for the length-32768 recurrence (memory-bound,
//    coalesced; ~268 MB of Bu traffic -> ~11us at 23.3 TB/s dominates runtime).

typedef float v2f __attribute__((ext_vector_type(2)));
typedef float v8f __attribute__((ext_vector_type(8)));
typedef unsigned int u32x4 __attribute__((ext_vector_type(4)));
typedef int i32x4 __attribute__((ext_vector_type(4)));
typedef int i32x8 __attribute__((ext_vector_type(8)));

#define L_SEQ  32768
#define DM     256       // D_MODEL
#define DH     256       // D_HIDDEN
#define CHUNK  256
#define NCHUNK (L_SEQ / CHUNK)   // 128
#define BPAD   260       // 256 + 4-dword TDM pad -> conflict-free b64 LDS reads

// ---------------------------------------------------------------------------
// TDM 2D tile load: build D# groups per CDNA5 ISA ch.8 and issue
// tensor_load_to_lds (6-arg builtin on this toolchain: g0, g1, g2, g3,
// extra group, cpol). data_size=4B, pad_enable with pad_interval=256 dwords /
// pad_amount=4 dwords -> LDS row stride 260 floats.
// ---------------------------------------------------------------------------
__device__ __forceinline__ void tdm_load_tile_2d(unsigned lds_off,
                                                 const float* gptr,
                                                 unsigned tile_d0, unsigned tile_d1,
                                                 unsigned tensor_d0, unsigned tensor_d1,
                                                 unsigned long long stride_d0) {
  const unsigned long long ga = (unsigned long long)(uintptr_t)gptr;
  u32x4 g0;
  g0[0] = 1u;                                   // count=1 (valid user D#)
  g0[1] = lds_off;                              // lds_addr (byte offset)
  g0[2] = (unsigned)ga;                         // global_addr[31:0]
  g0[3] = (unsigned)(ga >> 32) | (2u << 30);    // global_addr[56:32] | type=2
  i32x8 g1;
  g1[0] = (int)((2u << 16)        // data_size = 4 bytes
              | (1u << 20)        // pad_enable
              | (7u << 22)        // pad_interval = 256 dwords
              | (3u << 25));      // pad_amount  = 4 dwords
  g1[1] = (int)((tensor_d0 & 0xFFFFu) << 16);
  g1[2] = (int)(((tensor_d0 >> 16) & 0xFFFFu) | ((tensor_d1 & 0xFFFFu) << 16));
  g1[3] = (int)(((tensor_d1 >> 16) & 0xFFFFu) | ((tile_d0 & 0xFFFFu) << 16));
  g1[4] = (int)(tile_d1 & 0xFFFFu);             // tile_dim1 | tile_dim2=0
  g1[5] = (int)(unsigned)(stride_d0 & 0xFFFFFFFFull);
  g1[6] = (int)(unsigned)((stride_d0 >> 32) & 0xFFFFull);
  g1[7] = 0;
  i32x4 z4 = {0, 0, 0, 0};                       // 2D tensor: groups 2/3 unused
  i32x8 z8 = {0, 0, 0, 0, 0, 0, 0, 0};
  __builtin_amdgcn_tensor_load_to_lds(g0, g1, z4, z4, z8, 0);
}

// ---------------------------------------------------------------------------
// Phase 0: parameter packing.
//   lam = exp(-exp(nu)) * (cos(exp(theta)) + i sin(exp(theta)))
//   Bn  = (B_re + i B_im) * exp(gamma_log)[h]
//   CimN = -C_im   (f32 WMMA has no B-negate modifier; pre-negate)
// ---------------------------------------------------------------------------
__global__ void lru_pack(const float* __restrict__ nu_log,
                         const float* __restrict__ theta_log,
                         const float* __restrict__ gamma_log,
                         const float* __restrict__ B_re,
                         const float* __restrict__ B_im,
                         const float* __restrict__ C_im,
                         float* __restrict__ lamRe, float* __restrict__ lamIm,
                         float* __restrict__ BnRe,  float* __restrict__ BnIm,
                         float* __restrict__ CimN) {
  const int idx = blockIdx.x * blockDim.x + threadIdx.x;   // 0 .. DH*DM-1
  const int h   = idx >> 8;                                // B is (H, M) row-major
  const float g = expf(gamma_log[h]);
  BnRe[idx] = B_re[idx] * g;
  BnIm[idx] = B_im[idx] * g;
  CimN[idx] = -C_im[idx];                                  // C is (M, H) row-major
  if (idx < DH) {
    const float mag = expf(-expf(nu_log[idx]));
    const float th  = expf(theta_log[idx]);
    lamRe[idx] = mag * cosf(th);
    lamIm[idx] = mag * sinf(th);
  }
}

// ---------------------------------------------------------------------------
// Phase 1: Bu = X @ Bn^T (complex), via f32 WMMA 16x16x4.
// Block = 8 waves = 128 rows x 64 cols; wave owns a 16x64 tile (re+im accs).
// B panel (both planes) staged in LDS by TDM; A streamed from global with a
// one-step register pipeline. grid = (4 col groups fastest, 256 row blocks).
// Fragment layouts per CDNA5 ISA 7.12.2:
//   A 16x4:  lane = {row = lane&15, K = k0 + 2*(lane>>4)}, 2 VGPRs = K,K+1
//   B 4x16:  lane = {col = lane&15, K = k0 + 2*(lane>>4)}, 2 VGPRs = K,K+1
//   C 16x16: VGPR v = row v + 8*(lane>>4), col = lane&15
// ---------------------------------------------------------------------------
__global__ void __launch_bounds__(256) lru_bu_gemm(const float* __restrict__ X,
                                                   const float* __restrict__ BnRe,
                                                   const float* __restrict__ BnIm,
                                                   float* __restrict__ BuRe,
                                                   float* __restrict__ BuIm) {
  __shared__ float sB[2][64 * BPAD];              // ~130 KB of the 320 KB WGP LDS
  const int lane  = threadIdx.x & 31;
  const int wave  = threadIdx.x >> 5;
  const int col0  = blockIdx.x * 64;              // hidden channels (x fastest)
  const int row0  = (blockIdx.y * 8 + wave) * 16; // time rows
  const int mrow  = lane & 15;
  const int khalf = (lane >> 4) << 1;

  if (threadIdx.x == 0) {                         // TDM ignores EXEC; one issue/block
    tdm_load_tile_2d((unsigned)(uintptr_t)&sB[0][0], BnRe + (size_t)col0 * DM,
                     DM, 64, DM, 64, DM);
    tdm_load_tile_2d((unsigned)(uintptr_t)&sB[1][0], BnIm + (size_t)col0 * DM,
                     DM, 64, DM, 64, DM);
    __builtin_amdgcn_s_wait_tensorcnt(0);
  }
  __syncthreads();

  const float* arow = X + (size_t)(row0 + mrow) * DM;
  v8f accRe[4] = {}, accIm[4] = {};
  v2f a = *(const v2f*)(arow + khalf);
  for (int k0 = 0; k0 < DM; k0 += 4) {
    const v2f a_cur = a;
    if (k0 + 4 < DM) {
      __builtin_prefetch(arow + k0 + 68, 0, 3);   // global_prefetch_b8
      a = *(const v2f*)(arow + k0 + 4 + khalf);   // pipeline next A fragment
    }
#pragma unroll
    for (int t = 0; t < 4; ++t) {
      const int nl = t * 16 + mrow;               // panel-local column
      const v2f bre = *(const v2f*)(&sB[0][nl * BPAD + k0 + khalf]);
      const v2f bim = *(const v2f*)(&sB[1][nl * BPAD + k0 + khalf]);
      accRe[t] = __builtin_amdgcn_wmma_f32_16x16x4_f32(false, a_cur, false, bre,
                                                       (short)0, accRe[t], false, false);
      accIm[t] = __builtin_amdgcn_wmma_f32_16x16x4_f32(false, a_cur, false, bim,
                                                       (short)0, accIm[t], false, false);
    }
  }

  const int rAdd = (lane >> 4) ? 8 : 0;
#pragma unroll
  for (int t = 0; t < 4; ++t) {
    const int c = col0 + t * 16 + mrow;
#pragma unroll
    for (int v = 0; v < 8; ++v) {
      const size_t o = (size_t)(row0 + v + rAdd) * DH + c;
      BuRe[o] = accRe[t][v];
      BuIm[o] = accIm[t][v];
    }
  }
}

// ---------------------------------------------------------------------------
// Phase 2a: local scan inside each 256-step chunk (zero initial state),
// in place over Bu; emit per-chunk carry. thread = channel -> coalesced.
// ---------------------------------------------------------------------------
__global__ void __launch_bounds__(256) lru_scan_local(const float* __restrict__ lamRe,
                                                      const float* __restrict__ lamIm,
                                                      float* __restrict__ BuRe,
                                                      float* __restrict__ BuIm,
                                                      float* __restrict__ carryRe,
                                                      float* __restrict__ carryIm) {
  const int h = threadIdx.x;
  const int c = blockIdx.x;
  const float lr = lamRe[h], li = lamIm[h];
  float hr = 0.f, hi = 0.f;
  const size_t base = (size_t)c * CHUNK * DH + h;
  for (int t = 0; t < CHUNK; ++t) {
    const size_t idx = base + (size_t)t * DH;
    const float br = BuRe[idx], bi = BuIm[idx];
    const float nr = lr * hr - li * hi + br;
    const float ni = lr * hi + li * hr + bi;
    hr = nr; hi = ni;
    BuRe[idx] = hr; BuIm[idx] = hi;
  }
  carryRe[c * DH + h] = hr;
  carryIm[c * DH + h] = hi;
}

// ---------------------------------------------------------------------------
// Phase 2b: scan over the 128 chunk carries per channel.
// prefix[c] = state entering chunk c; lam^CHUNK by 8 complex squarings.
// ---------------------------------------------------------------------------
__global__ void __launch_bounds__(256) lru_scan_carry(const float* __restrict__ lamRe,
                                                      const float* __restrict__ lamIm,
                                                      const float* __restrict__ carryRe,
                                                      const float* __restrict__ carryIm,
                                                      float* __restrict__ prefixRe,
                                                      float* __restrict__ prefixIm) {
  const int h = threadIdx.x;
  float ar = lamRe[h], ai = lamIm[h];
#pragma unroll
  for (int i = 0; i < 8; ++i) {                 // lambda^(2^8) = lambda^CHUNK
    const float nr = ar * ar - ai * ai;
    const float ni = 2.f * ar * ai;
    ar = nr; ai = ni;
  }
  float sr = 0.f, si = 0.f;
  for (int c = 0; c < NCHUNK; ++c) {
    prefixRe[c * DH + h] = sr;
    prefixIm[c * DH + h] = si;
    const float cr = carryRe[c * DH + h], ci = carryIm[c * DH + h];
    const float nr = ar * sr - ai * si + cr;
    const float ni = ar * si + ai * sr + ci;
    sr = nr; si = ni;
  }
}

// ---------------------------------------------------------------------------
// Phase 2c: h_t = h_local_t + lambda^(t+1) * prefix[chunk], in place.
// ---------------------------------------------------------------------------
__global__ void __launch_bounds__(256) lru_apply_prefix(const float* __restrict__ lamRe,
                                                        const float* __restrict__ lamIm,
                                                        const float* __restrict__ prefixRe,
                                                        const float* __restrict__ prefixIm,
                                                        float* __restrict__ BuRe,
                                                        float* __restrict__ BuIm) {
  const int h = threadIdx.x;
  const int c = blockIdx.x;
  const float lr = lamRe[h], li = lamIm[h];
  const float sr = prefixRe[c * DH + h], si = prefixIm[c * DH + h];
  float pr = lr * sr - li * si;                 // lambda^1 * prefix
  float pi = lr * si + li * sr;
  const size_t base = (size_t)c * CHUNK * DH + h;
  for (int t = 0; t < CHUNK; ++t) {
    const size_t idx = base + (size_t)t * DH;
    BuRe[idx] += pr;
    BuIm[idx] += pi;
    const float nr = lr * pr - li * pi;         // advance lambda power
    const float ni = lr * pi + li * pr;
    pr = nr; pi = ni;
  }
}

// ---------------------------------------------------------------------------
// Phase 3: y = h_re @ C_re^T + h_im @ (-C_im)^T + X .* D.
// Same tiling as phase 1; C panels TDM-staged in LDS.
// ---------------------------------------------------------------------------
__global__ void __launch_bounds__(256) lru_out_gemm(const float* __restrict__ HRe,
                                                    const float* __restrict__ HIm,
                                                    const float* __restrict__ C_re,
                                                    const float* __restrict__ CimN,
                                                    const float* __restrict__ X,
                                                    const float* __restrict__ Dv,
                                                    float* __restrict__ Y) {
  __shared__ float sC[2][64 * BPAD];
  const int lane  = threadIdx.x & 31;
  const int wave  = threadIdx.x >> 5;
  const int col0  = blockIdx.x * 64;              // output features (x fastest)
  const int row0  = (blockIdx.y * 8 + wave) * 16; // time rows
  const int mrow  = lane & 15;
  const int khalf = (lane >> 4) << 1;

  if (threadIdx.x == 0) {
    tdm_load_tile_2d((unsigned)(uintptr_t)&sC[0][0], C_re + (size_t)col0 * DH,
                     DH, 64, DH, 64, DH);
    tdm_load_tile_2d((unsigned)(uintptr_t)&sC[1][0], CimN + (size_t)col0 * DH,
                     DH, 64, DH, 64, DH);
    __builtin_amdgcn_s_wait_tensorcnt(0);
  }
  __syncthreads();

  const float* hre_row = HRe + (size_t)(row0 + mrow) * DH;
  const float* him_row = HIm + (size_t)(row0 + mrow) * DH;
  v8f acc[4] = {};
  v2f are = *(const v2f*)(hre_row + khalf);
  v2f aim = *(const v2f*)(him_row + khalf);
  for (int k0 = 0; k0 < DH; k0 += 4) {
    const v2f are_c = are, aim_c = aim;
    if (k0 + 4 < DH) {
      __builtin_prefetch(hre_row + k0 + 68, 0, 3);
      __builtin_prefetch(him_row + k0 + 68, 0, 3);
      are = *(const v2f*)(hre_row + k0 + 4 + khalf);
      aim = *(const v2f*)(him_row + k0 + 4 + khalf);
    }
#pragma unroll
    for (int t = 0; t < 4; ++t) {
      const int nl = t * 16 + mrow;
      const v2f cre = *(const v2f*)(&sC[0][nl * BPAD + k0 + khalf]);
      const v2f cim = *(const v2f*)(&sC[1][nl * BPAD + k0 + khalf]);
      acc[t] = __builtin_amdgcn_wmma_f32_16x16x4_f32(false, are_c, false, cre,
                                                     (short)0, acc[t], false, false);
      acc[t] = __builtin_amdgcn_wmma_f32_16x16x4_f32(false, aim_c, false, cim,
                                                     (short)0, acc[t], false, false);
    }
  }

  const int rAdd = (lane >> 4) ? 8 : 0;
#pragma unroll
  for (int t = 0; t < 4; ++t) {
    const int c = col0 + t * 16 + mrow;
    const float d = Dv[c];
#pragma unroll
    for (int v = 0; v < 8; ++v) {
      const size_t o = (size_t)(row0 + v + rAdd) * DM + c;
      Y[o] = acc[t][v] + X[o] * d;
    }
  }
}

// ---------------------------------------------------------------------------
extern "C" void kernel_launch(void* const* d_in, const int* in_sizes, int n_in,
                              void* d_out, int out_size, void* d_ws, size_t ws_size,
                              hipStream_t stream) {
  const float* X         = (const float*)d_in[0];  // (L, M)
  const float* nu_log    = (const float*)d_in[1];  // (H,)
  const float* theta_log = (const float*)d_in[2];  // (H,)
  const float* gamma_log = (const float*)d_in[3];  // (H,)
  const float* B_re      = (const float*)d_in[4];  // (H, M)
  const float* B_im      = (const float*)d_in[5];  // (H, M)
  const float* C_re      = (const float*)d_in[6];  // (M, H)
  const float* C_im      = (const float*)d_in[7];  // (M, H)
  const float* Dv        = (const float*)d_in[8];  // (M,)
  float* Y = (float*)d_out;

  // Workspace layout (floats); total ~68 MB.
  float* w       = (float*)d_ws;
  float* lamRe   = w;                    w += DH;
  float* lamIm   = w;                    w += DH;
  float* BnRe    = w;                    w += DH * DM;
  float* BnIm    = w;                    w += DH * DM;
  float* CimN    = w;                    w += DM * DH;
  float* BuRe    = w;                    w += (size_t)L_SEQ * DH;
  float* BuIm    = w;                    w += (size_t)L_SEQ * DH;
  float* carryRe = w;                    w += NCHUNK * DH;
  float* carryIm = w;                    w += NCHUNK * DH;
  float* prefRe  = w;                    w += NCHUNK * DH;
  float* prefIm  = w;                    w += NCHUNK * DH;

  lru_pack<<<(DH * DM) / 256, 256, 0, stream>>>(nu_log, theta_log, gamma_log,
                                                B_re, B_im, C_im,
                                                lamRe, lamIm, BnRe, BnIm, CimN);

  dim3 gemmGrid(DM / 64, L_SEQ / 128);             // col groups fastest -> L2 reuse of A
  lru_bu_gemm<<<gemmGrid, 256, 0, stream>>>(X, BnRe, BnIm, BuRe, BuIm);

  lru_scan_local<<<NCHUNK, DH, 0, stream>>>(lamRe, lamIm, BuRe, BuIm, carryRe, carryIm);

  lru_scan_carry<<<1, DH, 0, stream>>>(lamRe, lamIm, carryRe, carryIm, prefRe, prefIm);

  lru_apply_prefix<<<NCHUNK, DH, 0, stream>>>(lamRe, lamIm, prefRe, prefIm, BuRe, BuIm);

  lru_out_gemm<<<gemmGrid, 256, 0, stream>>>(BuRe, BuIm, C_re, CimN, X, Dv, Y);
}